// Loss2_8461085573459
// MI455X (gfx1250) — compile-verified
//
#include <hip/hip_runtime.h>
#include <hip/hip_bf16.h>
#include <stdint.h>

#define NEGF (-1e30f)
#define KMAXC 200
#define SNUM (2 * KMAXC + 1)   // 401
#define TCHUNK 16
#define BLK2 416               // 13 waves, covers 401 states

typedef __attribute__((ext_vector_type(4))) unsigned int v4u;
typedef __attribute__((ext_vector_type(8))) int v8i;
typedef __attribute__((ext_vector_type(4))) int v4i;

// ---------------------------------------------------------------------------
// TDM: load a (tile_rows x KMAXC) float tile from global into LDS.
// D# packing per CDNA5 ISA §8 (group0: count/lds_addr/global_addr/type,
// group1: data_size=4B, tensor_dim0=K, tensor_dim1=rows_remaining,
// tile_dim0=K, tile_dim1=tile_rows, tensor_dim0_stride=K).
// 6-arg builtin form (clang-23 / therock headers): extra int32x8 group + cpol.
// ---------------------------------------------------------------------------
__device__ __forceinline__ void tdm_load_tile(const float* gsrc,
                                              uint32_t lds_byte_addr,
                                              int rows_remaining,
                                              int tile_rows) {
  const uint32_t cols = (uint32_t)KMAXC;
  uint64_t ga = (uint64_t)(uintptr_t)gsrc;

  v4u g0;
  g0[0] = 1u;                                   // count=1, is_restore=0, no gather
  g0[1] = lds_byte_addr;                        // lds_addr (bytes)
  g0[2] = (uint32_t)(ga & 0xFFFFFFFFu);         // global_addr[31:0]
  g0[3] = (uint32_t)((ga >> 32) & 0x1FFFFFFu)   // global_addr[56:32]
          | (2u << 30);                         // type=2 ("image")

  v8i g1;
  g1[0] = (int)(2u << 16);                      // data_size=2 -> 4 bytes; mask=0
  g1[1] = (int)((cols & 0xFFFFu) << 16);        // tensor_dim0[15:0] @ bits63:48
  g1[2] = (int)((cols >> 16) |
                (((uint32_t)rows_remaining & 0xFFFFu) << 16)); // tdim0 hi | tdim1 lo
  g1[3] = (int)(((uint32_t)rows_remaining >> 16) |
                ((cols & 0xFFFFu) << 16));      // tdim1 hi | tile_dim0
  g1[4] = (int)((uint32_t)tile_rows & 0xFFFFu); // tile_dim1; tile_dim2=0
  g1[5] = (int)cols;                            // tensor_dim0_stride[31:0]
  g1[6] = 0;                                    // stride hi | dim1_stride lo
  g1[7] = 0;

  v4i g2; g2[0] = 0; g2[1] = 0; g2[2] = 0; g2[3] = 0;
  v4i g3; g3[0] = 0; g3[1] = 0; g3[2] = 0; g3[3] = 0;
  v8i g4; g4[0] = 0; g4[1] = 0; g4[2] = 0; g4[3] = 0;
          g4[4] = 0; g4[5] = 0; g4[6] = 0; g4[7] = 0;

  __builtin_amdgcn_tensor_load_to_lds(g0, g1, g2, g3, g4, 0);
}

// ---------------------------------------------------------------------------
// Phase 1: per-(b,t) masked logsumexp over {blank=-1.0} U {attn[b,t,0..il-1]}.
// Grid: (ceil(T/8), B); one wave32 per t-row; shuffle reductions.
// (2-D grid avoids the 64-bit divide the flattened version emitted.)
// ---------------------------------------------------------------------------
__global__ __launch_bounds__(256) void row_lse_kernel(
    const float* __restrict__ attn, const int* __restrict__ in_lens,
    float* __restrict__ lse, int T) {
  const int lane = threadIdx.x & 31;
  const int w = threadIdx.x >> 5;
  const int b = blockIdx.y;
  const int t = blockIdx.x * 8 + w;
  if (t >= T) return;
  const int il = in_lens[b];
  const float* p = attn + ((size_t)b * T + t) * KMAXC;

  float v[7];
  float m = -1.0f;  // blank contributes to the max
#pragma unroll
  for (int i = 0; i < 7; ++i) {
    int j = lane + 32 * i;
    v[i] = (j < il) ? p[j] : NEGF;   // il <= KMAXC so j<il implies in-bounds
    m = fmaxf(m, v[i]);
  }
#pragma unroll
  for (int o = 16; o >= 1; o >>= 1) m = fmaxf(m, __shfl_xor(m, o));

  float s = 0.0f;
#pragma unroll
  for (int i = 0; i < 7; ++i) s += __expf(v[i] - m);
#pragma unroll
  for (int o = 16; o >= 1; o >>= 1) s += __shfl_xor(s, o);
  s += __expf(-1.0f - m);            // blank, counted once

  if (lane == 0) lse[(size_t)b * T + t] = m + __logf(s);
}

// ---------------------------------------------------------------------------
// Phase 2: CTC forward recursion. One block per batch element.
// TDM double-buffers TCHUNK x K emit tiles into LDS; lse chunk staged into
// LDS once per chunk; alpha double-buffered in LDS; one barrier per step.
// ---------------------------------------------------------------------------
__global__ __launch_bounds__(BLK2) void ctc_fwd_kernel(
    const float* __restrict__ attn, const int* __restrict__ in_lens,
    const int* __restrict__ out_lens, const float* __restrict__ lse,
    float* __restrict__ res, int T) {
  const int b = blockIdx.x;
  const int tid = threadIdx.x;

  __shared__ __align__(16) float tileA[TCHUNK * KMAXC];
  __shared__ __align__(16) float tileB[TCHUNK * KMAXC];
  __shared__ float alphaBuf[2][SNUM + 1];
  __shared__ float lseSh[TCHUNK];
  __shared__ float resSh;

  const int il = in_lens[b];
  const int ol = out_lens[b];
  const float* attnB = attn + (size_t)b * T * KMAXC;
  const float* lseB = lse + (size_t)b * T;

  if (tid == 0) resSh = NEGF;

  const int sid = tid;
  const bool isState = (sid < SNUM);
  const bool odd = (sid & 1) != 0;
  const int jcol = (sid - 1) >> 1;                 // key column for odd states
  const bool valid = isState && (sid <= 2 * il);
  const int end1 = 2 * il - 1, end2 = 2 * il;
  const bool wave0 = (tid < 32);

  const int nchunks = (T + TCHUNK - 1) / TCHUNK;
  const uint32_t ldsA = (uint32_t)(uintptr_t)&tileA[0];
  const uint32_t ldsB = (uint32_t)(uintptr_t)&tileB[0];

  if (wave0) {
    tdm_load_tile(attnB, ldsA, T, TCHUNK);
    if (nchunks > 1)
      tdm_load_tile(attnB + (size_t)TCHUNK * KMAXC, ldsB, T - TCHUNK, TCHUNK);
  }

  int cur = 0;
  int t = 0;
  for (int c = 0; c < nchunks; ++c) {
    const int t0 = c * TCHUNK;
    const int tend = min(TCHUNK, T - t0);

    // Stage this chunk's lse values into LDS (independent of the TDM tile).
    if (tid < tend) lseSh[tid] = lseB[t0 + tid];

    if (wave0) {
      if (c + 1 < nchunks) __builtin_amdgcn_s_wait_tensorcnt(1);
      else                 __builtin_amdgcn_s_wait_tensorcnt(0);
    }
    __syncthreads();  // tile + lse for chunk c resident in LDS for all waves

    const float* tile = (c & 1) ? tileB : tileA;

    for (int tt = 0; tt < tend; ++tt, ++t) {
      if (isState) {
        const float lse_t = lseSh[tt];
        float emit = NEGF;
        if (valid) {
          const float raw = odd ? tile[tt * KMAXC + jcol] : -1.0f;
          emit = raw - lse_t;
        }
        float na;
        if (t == 0) {
          na = (sid < 2) ? emit : NEGF;
        } else {
          const float a  = alphaBuf[cur][sid];
          const float p1 = (sid >= 1) ? alphaBuf[cur][sid - 1] : NEGF;
          const float p2 = (odd && sid >= 2) ? alphaBuf[cur][sid - 2] : NEGF;
          const float m = fmaxf(fmaxf(a, p1), p2);
          const float ssum = __expf(a - m) + __expf(p1 - m) + __expf(p2 - m);
          na = emit + m + __logf(ssum);
          na = fmaxf(na, NEGF);
        }
        alphaBuf[cur ^ 1][sid] = na;
      }
      __syncthreads();
      cur ^= 1;
      if (tid == 0 && t == ol - 1) {
        const float a1 = alphaBuf[cur][end1];
        const float a2 = alphaBuf[cur][end2];
        const float m = fmaxf(a1, a2);
        resSh = m + log1pf(__expf(fminf(a1, a2) - m));
      }
    }

    // All waves are past the last read of chunk c's buffer -> refill it.
    if (wave0 && (c + 2) < nchunks) {
      const float* gsrc = attnB + (size_t)(c + 2) * TCHUNK * KMAXC;
      tdm_load_tile(gsrc, (c & 1) ? ldsB : ldsA, T - (c + 2) * TCHUNK, TCHUNK);
    }
  }

  __syncthreads();
  if (tid == 0) res[b] = resSh;
}

// ---------------------------------------------------------------------------
// Phase 3: mean( -res[b] / in_lens[b] ). Single wave, deterministic.
// ---------------------------------------------------------------------------
__global__ __launch_bounds__(32) void finalize_kernel(
    const float* __restrict__ res, const int* __restrict__ in_lens,
    float* __restrict__ out, int B) {
  float s = 0.0f;
  for (int b = threadIdx.x; b < B; b += 32)
    s += -res[b] / (float)in_lens[b];
#pragma unroll
  for (int o = 16; o >= 1; o >>= 1) s += __shfl_xor(s, o);
  if (threadIdx.x == 0) out[0] = s / (float)B;
}

extern "C" void kernel_launch(void* const* d_in, const int* in_sizes, int n_in,
                              void* d_out, int out_size, void* d_ws, size_t ws_size,
                              hipStream_t stream) {
  const float* attn = (const float*)d_in[0];
  const int* in_lens = (const int*)d_in[1];
  const int* out_lens = (const int*)d_in[2];

  const int B = in_sizes[1];
  const int T = in_sizes[0] / (B * KMAXC);

  float* lse = (float*)d_ws;                  // B*T floats
  float* res = lse + (size_t)B * T;           // B floats

  dim3 grid1((T + 7) / 8, B, 1);
  row_lse_kernel<<<grid1, 256, 0, stream>>>(attn, in_lens, lse, T);
  ctc_fwd_kernel<<<B, BLK2, 0, stream>>>(attn, in_lens, out_lens, lse, res, T);
  finalize_kernel<<<1, 32, 0, stream>>>(res, in_lens, (float*)d_out, B);
}